// MLPBaselineFPProt_16295105921286
// MI455X (gfx1250) — compile-verified
//
#include <hip/hip_runtime.h>

// ---------------------------------------------------------------------------
// MLPBaselineFPProt on MI455X (gfx1250, wave32, WMMA, double-buffered LDS GEMM)
//
//   1) prot_sum:   E1[d,:] = sum_unique W1[1024+prot,:]   (dedup == .set)
//                  E2[d,:] = sum_unique W1[18049+prot,:]
//   2) WMMA GEMM:  E1 += x_drugs @ W1[0:1024,:]        (4000x1024x2048 f32)
//                  E2 += x_drugs @ W1[17025:18049,:]
//   3) assemble:   H1[b,:] = relu(E1[d1]+E2[d2]+a0*W1[17024]+a1*W1[34049]+b1)
//   4) WMMA GEMM:  H2 = relu(H1 @ W2 + b2)             (8192x2048x1024 f32)
//   5) final dot:  out[b] = H2[b,:] . W3 + b3
// ---------------------------------------------------------------------------

typedef __attribute__((ext_vector_type(2))) float v2f;
typedef __attribute__((ext_vector_type(4))) float v4f;
typedef __attribute__((ext_vector_type(8))) float v8f;

#define N_DRUGS    4000
#define FP_DIM     1024
#define N_PROTS    16000
#define T_PER_DRUG 8
#define BATCH      8192
#define H1DIM      2048
#define H2DIM      1024

// W1 row-offsets (row-major, ld = H1DIM)
#define ROW_FP1   0
#define ROW_PROT1 1024
#define ROW_ATTR1 17024
#define ROW_FP2   17025
#define ROW_PROT2 18049
#define ROW_ATTR2 34049

// GEMM tiling
#define BLK_M 32            // block rows   (2 waves in M)
#define BLK_N 256           // block cols   (4 waves in N, 64 cols per wave)
#define KB    32            // K-chunk staged in LDS
#define AS    (KB + 4)      // LDS row stride for A   (36: bank-spread + 16B align)
#define BS    (KB + 4)      // LDS col stride for B^T (36)

// ---------------------------------------------------------------------------
// 1) Per-drug protein row sums with dedup (multi-hot built via .set -> dup
//    protein ids contribute only once).
// ---------------------------------------------------------------------------
__global__ __launch_bounds__(256)
void prot_sum_kernel(const float* __restrict__ W1,
                     const int*   __restrict__ drug2prot,
                     float* __restrict__ E1,
                     float* __restrict__ E2) {
    const int d = blockIdx.y;
    const int j = blockIdx.x * blockDim.x + threadIdx.x;

    int ids[T_PER_DRUG];
#pragma unroll
    for (int t = 0; t < T_PER_DRUG; ++t)
        ids[t] = drug2prot[d * T_PER_DRUG + t];

    float s1 = 0.f, s2 = 0.f;
#pragma unroll
    for (int t = 0; t < T_PER_DRUG; ++t) {
        bool dup = false;
#pragma unroll
        for (int u = 0; u < t; ++u)
            dup = dup || (ids[u] == ids[t]);
        if (!dup) {
            s1 += W1[(size_t)(ROW_PROT1 + ids[t]) * H1DIM + j];
            s2 += W1[(size_t)(ROW_PROT2 + ids[t]) * H1DIM + j];
        }
    }
    E1[(size_t)d * H1DIM + j] = s1;
    E2[(size_t)d * H1DIM + j] = s2;
}

// ---------------------------------------------------------------------------
// 2/4) fp32 WMMA GEMM, double-buffered LDS, 2-stage software pipeline:
//      C[M,N] (+)= A[M,K] @ B[K,N]  (+ bias, optional relu)
//
//  grid = (N/BLK_N, M/BLK_M), block = 256 threads = 8 waves (2 M x 4 N).
//  Pipeline per K-chunk: copy regs into LDS buffer buf, barrier, issue global loads
//  for chunk+1 into regs (latency hidden under compute), WMMA over that LDS buffer.
//  B is stored transposed in LDS so the (k,k+1) fragment pair is one b64
//  read; stride-36 padding keeps the 16-lane groups conflict-free.
//  Requires M%32==0, N%256==0, K%KB==0 (true for all call sites) -> no
//  guards, EXEC all-ones for WMMA/barrier legality.
// ---------------------------------------------------------------------------
__global__ __launch_bounds__(256)
void wmma_gemm_f32(const float* __restrict__ A, int lda,
                   const float* __restrict__ B, int ldb,
                   float* __restrict__ C, int ldc,
                   int K,
                   const float* __restrict__ bias,
                   int accumulate, int relu) {
    __shared__ float Asmem[2][BLK_M * AS];   // [buf][row*AS + k]
    __shared__ float Bsmem[2][BLK_N * BS];   // [buf][col*BS + k]  (transposed)

    const int tid    = threadIdx.x;
    const int lane   = tid & 31;
    const int waveId = tid >> 5;
    const int mw     = waveId >> 2;       // 0..1
    const int nw     = waveId & 3;        // 0..3

    const int l15   = lane & 15;
    const int khalf = lane >> 4;          // 0: K pair {0,1}; 1: {2,3}

    const int mbase = blockIdx.y * BLK_M;
    const int nbase = blockIdx.x * BLK_N;

    // cooperative-fill thread mappings (chunk-invariant)
    const int a_row = tid >> 3;           // 0..31
    const int a_kq  = (tid & 7) * 4;      // 0,4,..,28
    const float* a_gsrc = A + (size_t)(mbase + a_row) * lda + a_kq;

    const int b_row0 = tid >> 6;          // 0..3 (advances by 4/pass)
    const int b_colq = (tid & 63) * 4;
    const float* b_gsrc = B + (size_t)b_row0 * ldb + nbase + b_colq;

    // per-wave fragment offsets within a buffer
    const int a_foff = (mw * 16 + l15) * AS + 2 * khalf;
    const int b_foff = (nw * 64 + l15) * BS + 2 * khalf;

    v8f acc[4] = {};

    // ---- prologue: fetch chunk 0 into registers ----
    v4f a_r = *(const v4f*)(a_gsrc);
    v4f b_r[8];
#pragma unroll
    for (int p = 0; p < 8; ++p)
        b_r[p] = *(const v4f*)(b_gsrc + (size_t)(p * 4) * ldb);

    int buf = 0;
    for (int kk0 = 0; kk0 < K; kk0 += KB) {
        // ---- dump staged registers into LDS buffer buf ----
        *(v4f*)&Asmem[buf][a_row * AS + a_kq] = a_r;
#pragma unroll
        for (int p = 0; p < 8; ++p) {
            const int row = b_row0 + p * 4;
#pragma unroll
            for (int j = 0; j < 4; ++j)
                Bsmem[buf][(b_colq + j) * BS + row] = b_r[p][j];
        }
        __syncthreads();

        // ---- prefetch next chunk into registers (overlaps compute) ----
        const int knext = kk0 + KB;
        if (knext < K) {
            a_r = *(const v4f*)(a_gsrc + knext);
#pragma unroll
            for (int p = 0; p < 8; ++p)
                b_r[p] = *(const v4f*)(b_gsrc + (size_t)(knext + p * 4) * ldb);
        }

        // ---- compute on LDS buffer buf: 1 A b64 + 4 B b64 reads per 4 WMMAs ----
        const float* af = &Asmem[buf][a_foff];
        const float* bf = &Bsmem[buf][b_foff];
#pragma unroll
        for (int k = 0; k < KB; k += 4) {
            v2f a = *(const v2f*)(af + k);
#pragma unroll
            for (int t = 0; t < 4; ++t) {
                v2f b = *(const v2f*)(bf + t * 16 * BS + k);
                acc[t] = __builtin_amdgcn_wmma_f32_16x16x4_f32(
                    false, a, false, b, (short)0, acc[t], false, false);
            }
        }
        buf ^= 1;
    }

    // ---- epilogue: C/D layout -> element r in lane l: M = rbase+r, N = col
    const int rbase = mbase + mw * 16 + 8 * khalf;
    const int ncol  = nbase + nw * 64 + l15;
#pragma unroll
    for (int t = 0; t < 4; ++t) {
        const int col = ncol + t * 16;
        const float bv = bias ? bias[col] : 0.f;
        float* cptr = C + (size_t)rbase * ldc + col;
#pragma unroll
        for (int r = 0; r < 8; ++r) {
            float v = acc[t][r] + bv;
            if (accumulate) v += cptr[(size_t)r * ldc];
            if (relu)       v  = fmaxf(v, 0.f);
            cptr[(size_t)r * ldc] = v;
        }
    }
}

// ---------------------------------------------------------------------------
// 3) Gather per-drug embeddings + attr columns + bias, relu -> H1 (8192x2048)
// ---------------------------------------------------------------------------
__global__ __launch_bounds__(256)
void assemble_h1_kernel(const float* __restrict__ E1,
                        const float* __restrict__ E2,
                        const int*   __restrict__ drug_pairs,
                        const float* __restrict__ attrs,
                        const float* __restrict__ W1,
                        const float* __restrict__ b1,
                        float* __restrict__ H1) {
    const int b = blockIdx.y;
    const int j = blockIdx.x * blockDim.x + threadIdx.x;
    const int d1 = drug_pairs[2 * b];
    const int d2 = drug_pairs[2 * b + 1];
    const float a0 = attrs[2 * b];
    const float a1 = attrs[2 * b + 1];

    float v = E1[(size_t)d1 * H1DIM + j]
            + E2[(size_t)d2 * H1DIM + j]
            + a0 * W1[(size_t)ROW_ATTR1 * H1DIM + j]
            + a1 * W1[(size_t)ROW_ATTR2 * H1DIM + j]
            + b1[j];
    H1[(size_t)b * H1DIM + j] = fmaxf(v, 0.f);
}

// ---------------------------------------------------------------------------
// 5) out[b] = H2[b,:] . W3 + b3   (one wave32 per output row)
// ---------------------------------------------------------------------------
__global__ __launch_bounds__(256)
void final_dot_kernel(const float* __restrict__ H2,
                      const float* __restrict__ W3,
                      const float* __restrict__ b3,
                      float* __restrict__ out) {
    const int row  = blockIdx.x * 8 + (threadIdx.x >> 5);
    const int lane = threadIdx.x & 31;
    float s = 0.f;
    for (int k = lane; k < H2DIM; k += 32)
        s += H2[(size_t)row * H2DIM + k] * W3[k];
#pragma unroll
    for (int off = 16; off > 0; off >>= 1)
        s += __shfl_down(s, off, 32);
    if (lane == 0)
        out[row] = s + b3[0];
}

// ---------------------------------------------------------------------------
extern "C" void kernel_launch(void* const* d_in, const int* in_sizes, int n_in,
                              void* d_out, int out_size, void* d_ws, size_t ws_size,
                              hipStream_t stream) {
    const float* x_drugs    = (const float*)d_in[0];
    const float* attrs      = (const float*)d_in[1];
    const float* W1         = (const float*)d_in[2];
    const float* b1         = (const float*)d_in[3];
    const float* W2         = (const float*)d_in[4];
    const float* b2         = (const float*)d_in[5];
    const float* W3         = (const float*)d_in[6];
    const float* b3         = (const float*)d_in[7];
    const int*   drug_pairs = (const int*)d_in[8];
    const int*   drug2prot  = (const int*)d_in[9];
    float* out = (float*)d_out;

    // Workspace: E1(4000x2048) | E2(4000x2048) | H1(8192x2048) | H2(8192x1024)
    float* E1 = (float*)d_ws;
    float* E2 = E1 + (size_t)N_DRUGS * H1DIM;
    float* H1 = E2 + (size_t)N_DRUGS * H1DIM;
    float* H2 = H1 + (size_t)BATCH   * H1DIM;

    // 1) per-drug dedup'd protein-row sums initialize E1/E2
    prot_sum_kernel<<<dim3(H1DIM / 256, N_DRUGS), 256, 0, stream>>>(
        W1, drug2prot, E1, E2);

    // 2) E1 += x_drugs @ W1[fp1];  E2 += x_drugs @ W1[fp2]
    {
        dim3 grid(H1DIM / BLK_N, N_DRUGS / BLK_M);   // (8, 125)
        wmma_gemm_f32<<<grid, 256, 0, stream>>>(
            x_drugs, FP_DIM, W1 + (size_t)ROW_FP1 * H1DIM, H1DIM,
            E1, H1DIM, FP_DIM, nullptr, /*acc=*/1, /*relu=*/0);
        wmma_gemm_f32<<<grid, 256, 0, stream>>>(
            x_drugs, FP_DIM, W1 + (size_t)ROW_FP2 * H1DIM, H1DIM,
            E2, H1DIM, FP_DIM, nullptr, /*acc=*/1, /*relu=*/0);
    }

    // 3) H1 = relu(E1[d1] + E2[d2] + attr terms + b1)
    assemble_h1_kernel<<<dim3(H1DIM / 256, BATCH), 256, 0, stream>>>(
        E1, E2, drug_pairs, attrs, W1, b1, H1);

    // 4) H2 = relu(H1 @ W2 + b2)
    {
        dim3 grid(H2DIM / BLK_N, BATCH / BLK_M);     // (4, 256)
        wmma_gemm_f32<<<grid, 256, 0, stream>>>(
            H1, H1DIM, W2, H2DIM, H2, H2DIM,
            H1DIM, b2, /*acc=*/0, /*relu=*/1);
    }

    // 5) out = H2 @ W3 + b3
    final_dot_kernel<<<BATCH / 8, 256, 0, stream>>>(H2, W3, b3, out);
}